// PatchSampleNonlocalOneGroup_51384988729574
// MI455X (gfx1250) — compile-verified
//
#include <hip/hip_runtime.h>
#include <hip/hip_bf16.h>

typedef __attribute__((ext_vector_type(2))) float v2f;
typedef __attribute__((ext_vector_type(8))) float v8f;

#define PATCH   16
#define HALF    8
#define SHALF   20
#define NCAND   40
#define WINSZ   56          // NCAND + PATCH
#define KTOT    768         // 3 * 16 * 16
#define NM      1600        // NCAND * NCAND
#define NSEL    256
#define IMGRES  256

// ---------------------------------------------------------------------------
// Stage 1: candidate patch distances via fp32 WMMA.
//   dist^2[m] = Sum w^2 - 2*<w,s> + Sum s^2
// Two chained V_WMMA_F32_16X16X4_F32 accumulators per 16-candidate tile:
//   acc : A = window rows,            B = sample vector broadcast over N
//   accq: A = window rows squared,    B = all-ones          (-> Sum w^2)
// Both are exact fp32 (RNE), matching the reference's fp32 math.
// ---------------------------------------------------------------------------
__global__ __launch_bounds__(512) void dist_kernel(const float* __restrict__ feat0,
                                                   const int* __restrict__ sloc,
                                                   float* __restrict__ dist2) {
    __shared__ float lwin[3 * WINSZ * WINSZ];   // 37,632 B
    __shared__ float lsmp[KTOT];                //  3,072 B

    const int tid = threadIdx.x;
    const int lh = sloc[0];
    const int lw = sloc[1];

    // Stage window (top-left at lh-28, lw-28) and center sample patch to LDS.
    for (int idx = tid; idx < 3 * WINSZ * WINSZ; idx += 512) {
        int c   = idx / (WINSZ * WINSZ);
        int rem = idx - c * (WINSZ * WINSZ);
        int r   = rem / WINSZ;
        int col = rem - r * WINSZ;
        lwin[idx] = feat0[c * 65536 + (lh - SHALF - HALF + r) * IMGRES + (lw - SHALF - HALF + col)];
    }
    for (int idx = tid; idx < KTOT; idx += 512) {
        int c = idx >> 8, y = (idx >> 4) & 15, x = idx & 15;
        lsmp[idx] = feat0[c * 65536 + (lh - HALF + y) * IMGRES + (lw - HALF + x)];
    }
    __syncthreads();

    const int lane = tid & 31;
    const int wave = tid >> 5;
    const int gw   = blockIdx.x * 16 + wave;   // global wave id (4 blocks x 16 waves)
    const int mrel = lane & 15;                // A-matrix row (M) this lane feeds
    const int koff = (lane >> 4) << 1;         // lanes 0-15: K=+0,+1 ; lanes 16-31: K=+2,+3

    // Sum s^2 once per wave (exact, full-wave shuffle reduction).
    float ssp = 0.f;
    for (int k = lane; k < KTOT; k += 32) { float sv = lsmp[k]; ssp = fmaf(sv, sv, ssp); }
    for (int off = 16; off; off >>= 1) ssp += __shfl_xor(ssp, off, 32);
    const float ss2 = ssp;

    for (int t = gw; t < NM / 16; t += 64) {           // 100 tiles of 16 candidates
        const int m  = t * 16 + mrel;
        const int mi = m / NCAND;
        const int mj = m - mi * NCAND;

        v8f acc  = {0.f, 0.f, 0.f, 0.f, 0.f, 0.f, 0.f, 0.f};  // <w,s>
        v8f accq = {0.f, 0.f, 0.f, 0.f, 0.f, 0.f, 0.f, 0.f};  // <w,w>
        for (int k0 = 0; k0 < KTOT; k0 += 4) {
            const int ka = k0 + koff;                  // even, ka&15 <= 14 -> ka+1 same row
            const int c = ka >> 8, y = (ka >> 4) & 15, x = ka & 15;
            const float* p = &lwin[c * (WINSZ * WINSZ) + (mi + y) * WINSZ + (mj + x)];
            v2f a;  a.x  = p[0];         a.y  = p[1];        // A[M=mrel, K=ka..ka+1]
            v2f b;  b.x  = lsmp[ka];     b.y  = lsmp[ka+1];  // B[K,N] broadcast over N
            v2f aq; aq.x = a.x * a.x;    aq.y = a.y * a.y;   // A element-wise squared
            v2f one; one.x = 1.0f;       one.y = 1.0f;       // all-ones B
            acc  = __builtin_amdgcn_wmma_f32_16x16x4_f32(
                false, a,  false, b,   (short)0, acc,  false, false);
            accq = __builtin_amdgcn_wmma_f32_16x16x4_f32(
                false, aq, false, one, (short)0, accq, false, false);
        }

        // Writer lanes: N==0 columns -> lanes 0..7 hold M=0..7 (vgpr r=M),
        // lanes 16..23 hold M=8..15 (vgpr r=M-8). Same slot in both chains.
        if (((lane >> 3) & 1) == 0) {
            const int   mr   = (lane < 8) ? lane : (lane - 8);
            const float corr = ((const float*)&acc )[lane & 7];
            const float sw   = ((const float*)&accq)[lane & 7];
            dist2[t * 16 + mr] = sw - 2.0f * corr + ss2;
        }
    }
}

// ---------------------------------------------------------------------------
// Stage 2: exact stable top-256 (ascending dist, ties -> lower flat index),
// matching jax.lax.top_k(-dist) semantics. O(1600^2) rank computation.
// ---------------------------------------------------------------------------
__global__ __launch_bounds__(256) void rank_kernel(const float* __restrict__ dist2,
                                                   const int* __restrict__ sloc,
                                                   int* __restrict__ locs_ws,
                                                   int* __restrict__ out_ids) {
    __shared__ float d[NM];
    for (int m = threadIdx.x; m < NM; m += 256) d[m] = dist2[m];
    __syncthreads();

    const int lh = sloc[0];
    const int lw = sloc[1];
    for (int m = threadIdx.x; m < NM; m += 256) {
        const float dm = d[m];
        int rank = 0;
        for (int mp = 0; mp < NM; ++mp) {
            const float dp = d[mp];
            rank += (dp < dm) || ((dp == dm) && (mp < m));
        }
        if (rank < NSEL) {
            const int i = m / NCAND;
            const int j = m - i * NCAND;
            const int h = lh - SHALF + i;
            const int w = lw - SHALF + j;
            locs_ws[2 * rank]     = h;
            locs_ws[2 * rank + 1] = w;
            out_ids[2 * rank]     = h;   // int32 bits into the float d_out buffer
            out_ids[2 * rank + 1] = w;
        }
    }
}

// ---------------------------------------------------------------------------
// Stage 3: gather one spatial position per patch across C channels, L2-norm.
// One block per patch; channels strided across 256 threads.
// ---------------------------------------------------------------------------
__global__ __launch_bounds__(256) void gather_norm_kernel(const float* __restrict__ feat,
                                                          const int* __restrict__ locs,
                                                          float* __restrict__ out,
                                                          int C, int H, int W, int shift) {
    const int p = blockIdx.x;
    const int h = locs[2 * p]     >> shift;   // patch_ids * H // 256  (H = 256>>shift)
    const int w = locs[2 * p + 1] >> shift;
    const long base = (long)h * W + w;
    const long hw   = (long)H * W;

    float vals[4];
    float sum = 0.f;
    int nv = 0;
    for (int c = threadIdx.x; c < C; c += 256) {
        float v = feat[(long)c * hw + base];
        vals[nv++] = v;
        sum = fmaf(v, v, sum);
    }

    __shared__ float red[256];
    red[threadIdx.x] = sum;
    __syncthreads();
    for (int s = 128; s > 0; s >>= 1) {
        if (threadIdx.x < s) red[threadIdx.x] += red[threadIdx.x + s];
        __syncthreads();
    }
    const float scale = 1.0f / (sqrtf(red[0]) + 1e-7f);

    nv = 0;
    for (int c = threadIdx.x; c < C; c += 256) {
        out[(long)p * C + c] = vals[nv++] * scale;
    }
}

// ---------------------------------------------------------------------------
extern "C" void kernel_launch(void* const* d_in, const int* in_sizes, int n_in,
                              void* d_out, int out_size, void* d_ws, size_t ws_size,
                              hipStream_t stream) {
    const float* feat0 = (const float*)d_in[0];  // (1,3,256,256)
    const float* feat1 = (const float*)d_in[1];  // (1,256,256,256)
    const float* feat2 = (const float*)d_in[2];  // (1,512,128,128)
    const float* feat3 = (const float*)d_in[3];  // (1,1024,64,64)
    const int*   sloc  = (const int*)d_in[4];    // (2,)

    float* out = (float*)d_out;
    // Output layout (flat, return order): out0[256*3] out1[256*256]
    // out2[256*512] out3[256*1024] patch_ids[256*2 as int32 bits]
    float* out0 = out;
    float* out1 = out + 768;
    float* out2 = out + 66304;
    float* out3 = out + 197376;
    int*   oid  = (int*)(out + 459520);

    float* dist2 = (float*)d_ws;                             // 1600 floats
    int*   locs  = (int*)((char*)d_ws + NM * sizeof(float)); // 256*2 ints

    dist_kernel<<<4, 512, 0, stream>>>(feat0, sloc, dist2);
    rank_kernel<<<1, 256, 0, stream>>>(dist2, sloc, locs, oid);
    gather_norm_kernel<<<NSEL, 256, 0, stream>>>(feat0, locs, out0,    3, 256, 256, 0);
    gather_norm_kernel<<<NSEL, 256, 0, stream>>>(feat1, locs, out1,  256, 256, 256, 0);
    gather_norm_kernel<<<NSEL, 256, 0, stream>>>(feat2, locs, out2,  512, 128, 128, 1);
    gather_norm_kernel<<<NSEL, 256, 0, stream>>>(feat3, locs, out3, 1024,  64,  64, 2);
}